// ArmaNetwork_24670292149150
// MI455X (gfx1250) — compile-verified
//
#include <hip/hip_runtime.h>
#include <hip/hip_bf16.h>

// ---------------- problem constants (match reference) ----------------
#define NN   50000      // nodes
#define EE   800000     // edges
#define GG   64         // graphs
#define HH   64         // hidden = F_IN
#define KK   2          // ARMA stacks
#define OUTC 24         // output classes
#define NH   (NN * HH)  // node-feature plane

typedef __attribute__((ext_vector_type(2))) float v2f;
typedef __attribute__((ext_vector_type(8))) float v8f;

// -------- fast fp32 global atomic add (global_atomic_add_f32) --------
__device__ __forceinline__ void atomAddF(float* p, float v) {
    unsafeAtomicAdd(p, v);
}

// ---------------- utility kernels ----------------
__global__ void zero_f32(float* __restrict__ p, int n) {
    int i = blockIdx.x * blockDim.x + threadIdx.x;
    if (i < n) p[i] = 0.0f;
}

__global__ void degree_kernel(const int* __restrict__ dst, float* __restrict__ deg, int e_cnt) {
    int i = blockIdx.x * blockDim.x + threadIdx.x;
    if (i < e_cnt) atomAddF(&deg[dst[i]], 1.0f);
}

__global__ void dinv_kernel(const float* __restrict__ deg, float* __restrict__ dinv, int n) {
    int i = blockIdx.x * blockDim.x + threadIdx.x;
    if (i < n) {
        float d = deg[i];
        dinv[i] = (d > 0.0f) ? rsqrtf(d) : 0.0f;
    }
}

__global__ void norm_kernel(const int* __restrict__ src, const int* __restrict__ dst,
                            const float* __restrict__ dinv, float* __restrict__ norm, int e_cnt) {
    int i = blockIdx.x * blockDim.x + threadIdx.x;
    if (i < e_cnt) norm[i] = dinv[src[i]] * dinv[dst[i]];
}

// ---------------- fp32 WMMA GEMM:  Y[M,64] = X[M,64] @ W[64,64] ------
// Block = 128 threads (4 waves). Block owns 16 rows; wave w owns cols [16w,16w+16).
// Uses V_WMMA_F32_16X16X4_F32 (full fp32 precision), 16 K-steps of 4.
__global__ void gemm64_wmma(const float* __restrict__ X,
                            const float* __restrict__ W,
                            float* __restrict__ Y, int M) {
    const int lane = threadIdx.x & 31;
    const int wave = threadIdx.x >> 5;        // N tile: 0..3
    const int half = lane >> 4;               // K-pair selector per ISA A/B layout
    const int l    = lane & 15;
    const int m0   = blockIdx.x * 16;
    const int n0   = wave * 16;
    const int row  = m0 + l;                  // A: lanes 0-15 and 16-31 both hold M = lane%16

    v8f c = {0.f, 0.f, 0.f, 0.f, 0.f, 0.f, 0.f, 0.f};

    #pragma unroll
    for (int ks = 0; ks < 16; ++ks) {
        const int kb = ks * 4 + half * 2;     // half=0 -> K{0,1}; half=1 -> K{2,3}
        v2f a, b;
        a.x = X[row * 64 + kb];
        a.y = X[row * 64 + kb + 1];
        b.x = W[(kb    ) * 64 + n0 + l];
        b.y = W[(kb + 1) * 64 + n0 + l];
        c = __builtin_amdgcn_wmma_f32_16x16x4_f32(false, a, false, b,
                                                  (short)0, c, false, false);
    }
    // D layout: VGPR v, lanes 0-15 -> M=v, lanes 16-31 -> M=v+8, N=lane%16
    #pragma unroll
    for (int v = 0; v < 8; ++v) {
        Y[(m0 + v + half * 8) * 64 + n0 + l] = c[v];
    }
}

// ---------------- edge propagation: agg[dst] += out[src] * norm ------
// one thread per (edge, 4-feature chunk); 16 chunks cover H=64
__global__ void propagate(const float* __restrict__ feat,   // [N,64] (one k)
                          const int* __restrict__ src, const int* __restrict__ dst,
                          const float* __restrict__ norm,
                          float* __restrict__ agg, int e_cnt) {
    int gid = blockIdx.x * blockDim.x + threadIdx.x;
    if (gid >= e_cnt * 16) return;
    int e = gid >> 4;
    int c = gid & 15;
    float nv = norm[e];
    int s = src[e], d = dst[e];
    const float4 v = *(const float4*)(feat + s * 64 + c * 4);
    float* ap = agg + d * 64 + c * 4;
    atomAddF(ap + 0, v.x * nv);
    atomAddF(ap + 1, v.y * nv);
    atomAddF(ap + 2, v.z * nv);
    atomAddF(ap + 3, v.w * nv);
}

// ---------------- out = relu(agg + root + bias)  over [K,N,H] --------
__global__ void combine(const float* __restrict__ agg, const float* __restrict__ root,
                        const float* __restrict__ bias,  // [K,64]
                        float* __restrict__ out, int n) { // n = K*N*H
    int i = blockIdx.x * blockDim.x + threadIdx.x;
    if (i >= n) return;
    int k = i / NH;
    int h = i & 63;
    float v = agg[i] + root[i] + bias[k * 64 + h];
    out[i] = fmaxf(v, 0.0f);
}

// ---------------- h = mean over K (outer relu is identity on relu'd) -
__global__ void mean_k(const float* __restrict__ out, float* __restrict__ h, int n) {
    int i = blockIdx.x * blockDim.x + threadIdx.x;
    if (i < n) h[i] = 0.5f * (out[i] + out[i + NH]);
}

// ---------------- pooling ----------------
__global__ void count_kernel(const int* __restrict__ batch, float* __restrict__ counts, int n) {
    int i = blockIdx.x * blockDim.x + threadIdx.x;
    if (i < n) atomAddF(&counts[batch[i]], 1.0f);
}

__global__ void pool_accum(const float* __restrict__ h, const int* __restrict__ batch,
                           float* __restrict__ pooled, int n) {
    int gid = blockIdx.x * blockDim.x + threadIdx.x;
    if (gid >= n * 16) return;
    int node = gid >> 4;
    int c = gid & 15;
    int g = batch[node];
    const float4 v = *(const float4*)(h + node * 64 + c * 4);
    float* pp = pooled + g * 64 + c * 4;
    atomAddF(pp + 0, v.x);
    atomAddF(pp + 1, v.y);
    atomAddF(pp + 2, v.z);
    atomAddF(pp + 3, v.w);
}

// ---------------- head: out[g,o] = (pooled[g]/cnt) @ lin_w + lin_b ---
__global__ void final_gemm(const float* __restrict__ pooled, const float* __restrict__ counts,
                           const float* __restrict__ lin_w, const float* __restrict__ lin_b,
                           float* __restrict__ out) {
    int o = blockIdx.x;   // 0..23
    int g = threadIdx.x;  // 0..63
    float scale = 1.0f / fmaxf(counts[g], 1.0f);
    float s = 0.0f;
    #pragma unroll
    for (int hh = 0; hh < 64; ++hh) s += pooled[g * 64 + hh] * lin_w[hh * OUTC + o];
    out[g * OUTC + o] = s * scale + lin_b[o];
}

// ---------------- host side ----------------
static inline int cdiv(int a, int b) { return (a + b - 1) / b; }

static void run_layer(const float* xin,
                      const float* init_w, const float* w_mid,
                      const float* root_w, const float* bias,
                      const int* src, const int* dst, const float* norm,
                      float* out, float* root, float* agg, float* tmp, float* h,
                      hipStream_t stream) {
    const int gemm_blocks = NN / 16;  // 3125, exact
    // out0 = x @ init_w[k], root = x @ root_w[k]
    for (int k = 0; k < KK; ++k) {
        gemm64_wmma<<<gemm_blocks, 128, 0, stream>>>(xin, init_w + k * 4096, out  + k * NH, NN);
        gemm64_wmma<<<gemm_blocks, 128, 0, stream>>>(xin, root_w + k * 4096, root + k * NH, NN);
    }
    // t = 0
    zero_f32<<<cdiv(KK * NH, 256), 256, 0, stream>>>(agg, KK * NH);
    for (int k = 0; k < KK; ++k)
        propagate<<<cdiv(EE * 16, 256), 256, 0, stream>>>(out + k * NH, src, dst, norm, agg + k * NH, EE);
    combine<<<cdiv(KK * NH, 256), 256, 0, stream>>>(agg, root, bias, out, KK * NH);
    // t = 1: out = out @ w[k], then propagate again
    for (int k = 0; k < KK; ++k)
        gemm64_wmma<<<gemm_blocks, 128, 0, stream>>>(out + k * NH, w_mid + k * 4096, tmp + k * NH, NN);
    zero_f32<<<cdiv(KK * NH, 256), 256, 0, stream>>>(agg, KK * NH);
    for (int k = 0; k < KK; ++k)
        propagate<<<cdiv(EE * 16, 256), 256, 0, stream>>>(tmp + k * NH, src, dst, norm, agg + k * NH, EE);
    combine<<<cdiv(KK * NH, 256), 256, 0, stream>>>(agg, root, bias, out, KK * NH);
    // mean over K -> h
    mean_k<<<cdiv(NH, 256), 256, 0, stream>>>(out, h, NH);
}

extern "C" void kernel_launch(void* const* d_in, const int* in_sizes, int n_in,
                              void* d_out, int out_size, void* d_ws, size_t ws_size,
                              hipStream_t stream) {
    (void)in_sizes; (void)n_in; (void)out_size; (void)ws_size;

    const float* x      = (const float*)d_in[0];
    const int*   ei     = (const int*)d_in[1];   // [2,E]
    const int*   batch  = (const int*)d_in[2];
    const float* init_w[3] = { (const float*)d_in[3],  (const float*)d_in[7],  (const float*)d_in[11] };
    const float* w_mid [3] = { (const float*)d_in[4],  (const float*)d_in[8],  (const float*)d_in[12] };
    const float* root_w[3] = { (const float*)d_in[5],  (const float*)d_in[9],  (const float*)d_in[13] };
    const float* bias  [3] = { (const float*)d_in[6],  (const float*)d_in[10], (const float*)d_in[14] };
    const float* lin_w  = (const float*)d_in[15];
    const float* lin_b  = (const float*)d_in[16];
    float*       outp   = (float*)d_out;

    const int* src = ei;
    const int* dst = ei + EE;

    // ---- workspace carve-up (floats) ----
    float* ws = (float*)d_ws;
    size_t off = 0;
    float* deg    = ws + off; off += NN;
    float* dinv   = ws + off; off += NN;
    float* norm   = ws + off; off += EE;
    float* out    = ws + off; off += (size_t)KK * NH;
    float* root   = ws + off; off += (size_t)KK * NH;
    float* agg    = ws + off; off += (size_t)KK * NH;
    float* tmp    = ws + off; off += (size_t)KK * NH;
    float* h      = ws + off; off += NH;
    float* pooled = ws + off; off += GG * 64;
    float* counts = ws + off; off += GG;

    // ---- gcn norm ----
    zero_f32<<<cdiv(NN, 256), 256, 0, stream>>>(deg, NN);
    degree_kernel<<<cdiv(EE, 256), 256, 0, stream>>>(dst, deg, EE);
    dinv_kernel<<<cdiv(NN, 256), 256, 0, stream>>>(deg, dinv, NN);
    norm_kernel<<<cdiv(EE, 256), 256, 0, stream>>>(src, dst, dinv, norm, EE);

    // ---- three ARMA layers ----
    run_layer(x, init_w[0], w_mid[0], root_w[0], bias[0], src, dst, norm,
              out, root, agg, tmp, h, stream);
    run_layer(h, init_w[1], w_mid[1], root_w[1], bias[1], src, dst, norm,
              out, root, agg, tmp, h, stream);
    run_layer(h, init_w[2], w_mid[2], root_w[2], bias[2], src, dst, norm,
              out, root, agg, tmp, h, stream);

    // ---- global mean pool + linear head ----
    zero_f32<<<cdiv(GG * 64 + GG, 256), 256, 0, stream>>>(pooled, GG * 64 + GG); // pooled+counts contiguous
    count_kernel<<<cdiv(NN, 256), 256, 0, stream>>>(batch, counts, NN);
    pool_accum<<<cdiv(NN * 16, 256), 256, 0, stream>>>(h, batch, pooled, NN);
    final_gemm<<<OUTC, GG, 0, stream>>>(pooled, counts, lin_w, lin_b, outp);
}